// GAT_57913339019683
// MI455X (gfx1250) — compile-verified
//
#include <hip/hip_runtime.h>
#include <hip/hip_bf16.h>
#include <math.h>

// ---------------------------------------------------------------------------
// CRF NLL on gfx1250 (MI455X).
//
// forward recurrence  alpha'[j] = logsumexp_i(alpha[i] + trans[j,i]) + feat[j]
// is rewritten in probability space (scaled forward algorithm):
//     P' = diag(exp(feat_t)) * E * P,   E = exp(trans)
// which is a real 16x16 matmul -> V_WMMA_F32_16X16X4_F32 (4 chained k=4 WMMAs).
// T is split into S segments; each wave composes the 16x16 segment transfer
// matrix (associativity of the log-semiring map), giving B*S = 16384 waves.
// Per-column log-scales keep fp32 in range. A combine kernel chains the
// segment matrices in log space and subtracts the (fused) gold-path score.
// ---------------------------------------------------------------------------

#define CB 1024        // batch
#define CT 4096        // time
#define CK 16          // tags
#define CS 16          // segments over T
#define CTS (CT / CS)  // 256 steps per segment
#define C_START 14
#define C_STOP 15

typedef float v2f __attribute__((ext_vector_type(2)));
typedef float v8f __attribute__((ext_vector_type(8)));

// One wave (32 threads) per (batch, segment).
// A-operand layout (16x4 f32): lane l holds row M = l&15; VGPR v, half h=l>>4
//   covers k_local = v + 2h  -> global k = 4c + 2h + v for chunk c.
// B-operand assumed mirrored:  element (K = 4c+2h+v, N = l&15).
// C/D layout: lane l, VGPR v -> (M = v + 8h, N = l&15).
__global__ __launch_bounds__(32) void crf_fwd_seg(
    const float* __restrict__ feats,   // [B, T, K]
    const int*   __restrict__ tags,    // [B, T]
    const float* __restrict__ trans,   // [K, K]  trans[j][i] = score(i->j)
    float* __restrict__ segM,          // [B*S, 16, 16]  log segment matrices
    float* __restrict__ segGold)       // [B*S] gold partial scores
{
  __shared__ float ldsT[256];   // raw transitions (gold lookups)
  __shared__ float ldsP[256];   // state relayout buffer, [N][M]

  const int wid = blockIdx.x;         // 0 .. B*S-1
  const int b   = wid / CS;
  const int s   = wid % CS;
  const int l   = threadIdx.x;        // 0..31
  const int n   = l & 15;             // column / row index within tile
  const int h   = l >> 4;             // wave half

  // cooperative copy of transitions into LDS (8 floats per lane)
  {
    const float4* t4 = (const float4*)trans;
    ((float4*)ldsT)[l * 2 + 0] = t4[l * 2 + 0];
    ((float4*)ldsT)[l * 2 + 1] = t4[l * 2 + 1];
  }
  __syncthreads();

  // E = exp(trans) held in A-operand layout: lane owns row j = n.
  float eA[4][2];
#pragma unroll
  for (int c = 0; c < 4; ++c)
#pragma unroll
    for (int v = 0; v < 2; ++v)
      eA[c][v] = __expf(ldsT[n * 16 + (4 * c + 2 * h + v)]);

  // State P (B-operand layout) starts as identity; per-column log-scale cs.
  float pB[4][2];
#pragma unroll
  for (int c = 0; c < 4; ++c)
#pragma unroll
    for (int v = 0; v < 2; ++v)
      pB[c][v] = ((4 * c + 2 * h + v) == n) ? 1.0f : 0.0f;
  float cs = 0.0f;

  // gold-path partial score for this segment
  const int t0 = s * CTS;
  int tag_prev = (s == 0) ? C_START : tags[(size_t)b * CT + t0 - 1];
  float gold = 0.0f;

  const float* fb = feats + (size_t)b * CT * CK;
  const int*   tb = tags + (size_t)b * CT;

  for (int tt = 0; tt < CTS; tt += 4) {
    if (tt + 8 < CTS)  // speculative prefetch of the feat stream
      __builtin_prefetch(&fb[(size_t)(t0 + tt + 8) * CK + n], 0, 0);

#pragma unroll
    for (int u = 0; u < 4; ++u) {
      const int t = t0 + tt + u;

      // emission for tag n at time t (both wave halves load the same 64B line)
      const float fv = fb[(size_t)t * CK + n];
      const int tg = tb[t];

      // ---- gold score (uniform across lanes; lane 0 writes at the end) ----
      gold += __shfl(fv, tg, 32) + ldsT[tg * 16 + tag_prev];
      tag_prev = tg;

      // ---- P' = diag(exp(feat_t)) * E * P  via 4 chained k=4 WMMAs ----
      const float f = __expf(fv);   // lane's row scale: row j == lane row
      v8f acc = {0.f, 0.f, 0.f, 0.f, 0.f, 0.f, 0.f, 0.f};
#pragma unroll
      for (int c = 0; c < 4; ++c) {
        v2f av = {eA[c][0] * f, eA[c][1] * f};   // A chunk = diag(f)*E cols 4c..4c+3
        v2f bv = {pB[c][0], pB[c][1]};           // B chunk = P rows 4c..4c+3
        acc = __builtin_amdgcn_wmma_f32_16x16x4_f32(
            false, av, false, bv, (short)0, acc, false, false);
      }

      // ---- D -> B-operand relayout through LDS ([N][M] so stores coalesce) ----
      __syncthreads();
      {
        float4 s0 = {acc[0], acc[1], acc[2], acc[3]};
        float4 s1 = {acc[4], acc[5], acc[6], acc[7]};
        *(float4*)&ldsP[n * 16 + 8 * h + 0] = s0;   // ds_store_b128 x2
        *(float4*)&ldsP[n * 16 + 8 * h + 4] = s1;
      }
      __syncthreads();
#pragma unroll
      for (int c = 0; c < 4; ++c) {
        float2 ld = *(float2*)&ldsP[n * 16 + 4 * c + 2 * h];  // ds_load_b64 x4
        pB[c][0] = ld.x;
        pB[c][1] = ld.y;
      }
    }

    // ---- per-column renormalization every 4 steps ----
    float m = pB[0][0];
#pragma unroll
    for (int c = 0; c < 4; ++c)
#pragma unroll
      for (int v = 0; v < 2; ++v) m = fmaxf(m, pB[c][v]);
    m = fmaxf(m, __shfl_xor(m, 16, 32));   // combine the two halves of column n
    m = fmaxf(m, 1e-30f);                  // dead columns (e.g. i==STOP) stay 0
    const float inv = 1.0f / m;
#pragma unroll
    for (int c = 0; c < 4; ++c)
#pragma unroll
      for (int v = 0; v < 2; ++v) pB[c][v] *= inv;
    cs += __logf(m);
  }

  // ---- emit log segment matrix: logM[j][i] = log(P[j][i]) + cs[i] ----
  float* M = segM + (size_t)wid * 256;
#pragma unroll
  for (int c = 0; c < 4; ++c)
#pragma unroll
    for (int v = 0; v < 2; ++v) {
      const int k = 4 * c + 2 * h + v;             // row j
      M[k * 16 + n] = __logf(pB[c][v]) + cs;       // __logf(0) -> -inf ok
    }
  if (l == 0) segGold[wid] = gold;
}

// Chain the S segment matrices in log space per batch, add STOP row,
// subtract gold score. One thread per batch (tiny: 1024 * ~24K flops).
__global__ __launch_bounds__(256) void crf_combine(
    const float* __restrict__ segM, const float* __restrict__ segGold,
    const float* __restrict__ trans, const int* __restrict__ tags,
    float* __restrict__ out)
{
  const int b = blockIdx.x * blockDim.x + threadIdx.x;
  if (b >= CB) return;

  const float* Mb = segM + (size_t)b * CS * 256;

  float alpha[CK];
#pragma unroll
  for (int j = 0; j < CK; ++j) alpha[j] = Mb[j * 16 + C_START];  // M_0 * e_START

  for (int s = 1; s < CS; ++s) {
    const float* M = Mb + (size_t)s * 256;
    float na[CK];
#pragma unroll
    for (int j = 0; j < CK; ++j) {
      float mx = -INFINITY;
#pragma unroll
      for (int k = 0; k < CK; ++k) mx = fmaxf(mx, M[j * 16 + k] + alpha[k]);
      if (mx > -1e30f) {
        float sum = 0.0f;
#pragma unroll
        for (int k = 0; k < CK; ++k) sum += __expf(M[j * 16 + k] + alpha[k] - mx);
        na[j] = mx + __logf(sum);
      } else {
        na[j] = -INFINITY;
      }
    }
#pragma unroll
    for (int j = 0; j < CK; ++j) alpha[j] = na[j];
  }

  // forward score: logsumexp_j(alpha[j] + trans[STOP, j])
  float mx = -INFINITY;
#pragma unroll
  for (int j = 0; j < CK; ++j) mx = fmaxf(mx, alpha[j] + trans[C_STOP * 16 + j]);
  float sum = 0.0f;
#pragma unroll
  for (int j = 0; j < CK; ++j) sum += __expf(alpha[j] + trans[C_STOP * 16 + j] - mx);
  const float fwd = mx + __logf(sum);

  float gold = trans[C_STOP * 16 + tags[(size_t)b * CT + CT - 1]];
  for (int s = 0; s < CS; ++s) gold += segGold[b * CS + s];

  out[b] = fwd - gold;
}

extern "C" void kernel_launch(void* const* d_in, const int* in_sizes, int n_in,
                              void* d_out, int out_size, void* d_ws, size_t ws_size,
                              hipStream_t stream) {
  const float* feats = (const float*)d_in[0];   // [B,T,K] f32
  const int*   tags  = (const int*)d_in[1];     // [B,T] i32
  const float* trans = (const float*)d_in[2];   // [K,K] f32
  float* out = (float*)d_out;                   // [B] f32

  // workspace: segM [B*S*256] f32 (16.78 MB) then segGold [B*S] f32
  float* segM = (float*)d_ws;
  float* segGold = segM + (size_t)CB * CS * 256;

  crf_fwd_seg<<<CB * CS, 32, 0, stream>>>(feats, tags, trans, segM, segGold);
  crf_combine<<<(CB + 255) / 256, 256, 0, stream>>>(segM, segGold, trans, tags, out);
}